// SkeletonDecoder_72249939853648
// MI455X (gfx1250) — compile-verified
//
#include <hip/hip_runtime.h>

typedef __attribute__((ext_vector_type(2))) float v2f;
typedef __attribute__((ext_vector_type(4))) float v4f;
typedef __attribute__((ext_vector_type(8))) float v8f;

#define PRES 64
#define P 4096              // 64*64 pixels per plane
#define SB 128              // S*B = 2*64
#define NK 20
#define KEPS 1e-6f

// output segment offsets (in floats), segments concatenated in return order
#define O0 0ull                                         // fv_raw  (SB,40,P)
#define O1 ((unsigned long long)SB * 40 * P)            // fd_raw  (SB,20,P)
#define O2 (O1 + (unsigned long long)SB * 20 * P)       // bv_raw  (SB,40,P)
#define O3 (O2 + (unsigned long long)SB * 40 * P)       // bd_raw  (SB,20,P)
#define O4 (O3 + (unsigned long long)SB * 20 * P)       // m_raw   (SB,20,P)
#define OKPS (O4 + (unsigned long long)SB * 20 * P)     // kps     (SB,21,2)

// Blocks [0, SB*NK): fused copy (fv pair, fd, m planes) + reduction for one (sb,k).
// Blocks [SB*NK, SB*NK + SB*15): copy-only, 4 planes each of channels 60..119.
__global__ __launch_bounds__(256) void skel_fused(const float* __restrict__ x,
                                                  float* __restrict__ out,
                                                  float* __restrict__ ws) {
  const int bid = blockIdx.x;
  const int tid = threadIdx.x;

  if (bid < SB * NK) {
    const int k  = bid % NK;
    const int sb = bid / NK;
    const float* xb  = x + (size_t)sb * 140 * P;
    const float* fvp = xb + (size_t)(2 * k) * P;      // two planes, (v0,v1) interleaved
    const float* fdp = xb + (size_t)(40 + k) * P;
    const float* mp  = xb + (size_t)(120 + k) * P;
    float* ofv = out + O0 + ((size_t)sb * 40 + 2 * k) * P;
    float* ofd = out + O1 + ((size_t)sb * 20 + k) * P;
    float* om  = out + O4 + ((size_t)sb * 20 + k) * P;

    float aM = 0.f, aA0 = 0.f, aA1 = 0.f, aLi = 0.f, aLj = 0.f;
    #pragma unroll
    for (int r = 0; r < 4; ++r) {
      const int q = tid + 256 * r;                    // v4f chunk: pixels 4q..4q+3
      v4f m4  = __builtin_nontemporal_load(((const v4f*)mp)  + q);
      v4f d4  = __builtin_nontemporal_load(((const v4f*)fdp) + q);
      v4f v4a = __builtin_nontemporal_load(((const v4f*)fvp) + 2 * q);
      v4f v4b = __builtin_nontemporal_load(((const v4f*)fvp) + 2 * q + 1);
      __builtin_nontemporal_store(m4,  ((v4f*)om)  + q);
      __builtin_nontemporal_store(d4,  ((v4f*)ofd) + q);
      __builtin_nontemporal_store(v4a, ((v4f*)ofv) + 2 * q);
      __builtin_nontemporal_store(v4b, ((v4f*)ofv) + 2 * q + 1);

      const float mm[4] = {m4.x, m4.y, m4.z, m4.w};
      const float dd[4] = {d4.x, d4.y, d4.z, d4.w};
      const float v0[4] = {v4a.x, v4a.z, v4b.x, v4b.z};
      const float v1[4] = {v4a.y, v4a.w, v4b.y, v4b.w};
      #pragma unroll
      for (int u = 0; u < 4; ++u) {
        const int p = 4 * q + u;
        const float fi = (float)(p >> 6);
        const float fj = (float)(p & 63);
        const float md = mm[u] * dd[u] * 64.0f;       // m*fd*res
        aM  += mm[u];
        aA0 = fmaf(md, v0[u], aA0);
        aA1 = fmaf(md, v1[u], aA1);
        aLi = fmaf(mm[u], fi, aLi);
        aLj = fmaf(mm[u], fj, aLj);
      }
    }

    // intra-wave (wave32) butterfly reduction of all 5 accumulators
    #pragma unroll
    for (int off = 16; off > 0; off >>= 1) {
      aM  += __shfl_xor(aM,  off, 32);
      aA0 += __shfl_xor(aA0, off, 32);
      aA1 += __shfl_xor(aA1, off, 32);
      aLi += __shfl_xor(aLi, off, 32);
      aLj += __shfl_xor(aLj, off, 32);
    }

    __shared__ float wpart[8][16];                    // [wave][accum row], zero-padded
    const int lane = tid & 31;
    const int wv   = tid >> 5;
    if (tid < 128) ((float*)wpart)[tid] = 0.0f;
    __syncthreads();
    if (lane == 0) {
      wpart[wv][0] = aM;  wpart[wv][1] = aA0; wpart[wv][2] = aA1;
      wpart[wv][3] = aLi; wpart[wv][4] = aLj;
    }
    __syncthreads();

    if (wv == 0) {
      // Cross-wave reduction via V_WMMA_F32_16X16X4_F32 (exact):
      // A[m,k] = wpart[k][m] (m = accumulator row, k = wave), B = e0 column,
      // so D[m,0] = sum_k wpart[k][m]. Two accumulating WMMAs cover 8 waves.
      const int m  = lane & 15;
      const int hi = lane >> 4;                       // 0 -> K=0,1 ; 1 -> K=2,3
      v2f a, a2;
      a.x  = wpart[hi * 2 + 0][m];     a.y  = wpart[hi * 2 + 1][m];
      a2.x = wpart[4 + hi * 2 + 0][m]; a2.y = wpart[4 + hi * 2 + 1][m];
      const float be = (m == 0) ? 1.0f : 0.0f;        // B[k,0] = 1, else 0
      v2f b; b.x = be; b.y = be;
      v8f c = {};
      c = __builtin_amdgcn_wmma_f32_16x16x4_f32(false, a,  false, b, (short)0, c, false, false);
      c = __builtin_amdgcn_wmma_f32_16x16x4_f32(false, a2, false, b, (short)0, c, false, false);
      if (lane == 0) {
        const float M = c[0], A0 = c[1], A1 = c[2], Li = c[3], Lj = c[4];
        const float inv = 1.0f / (M + KEPS);
        float* w = ws + (size_t)bid * 4;
        w[0] = (A0 + Li) * inv;                       // vals.x
        w[1] = (A1 + Lj) * inv;                       // vals.y
        w[2] = (Li - A0) * inv;                       // wnum.x/den
        w[3] = (Lj - A1) * inv;                       // wnum.y/den
      }
    }
  } else {
    // copy-only: bv_raw / bd_raw channels 60..119, 4 planes per block
    const int idx = bid - SB * NK;
    const int j15 = idx % 15;
    const int sb  = idx / 15;
    const float* xb = x + (size_t)sb * 140 * P;
    #pragma unroll
    for (int u = 0; u < 4; ++u) {
      const int c = 60 + 4 * j15 + u;
      const float* src = xb + (size_t)c * P;
      float* dst = (c < 100) ? (out + O2 + ((size_t)sb * 40 + (c - 60)) * P)
                             : (out + O3 + ((size_t)sb * 20 + (c - 100)) * P);
      #pragma unroll
      for (int r = 0; r < 4; ++r) {
        const int q = tid + 256 * r;
        v4f v = __builtin_nontemporal_load(((const v4f*)src) + q);
        __builtin_nontemporal_store(v, ((v4f*)dst) + q);
      }
    }
  }
}

__global__ void skel_finalize(const float* __restrict__ ws, float* __restrict__ out) {
  const int sb = blockIdx.x * blockDim.x + threadIdx.x;
  if (sb >= SB) return;
  float* kp = out + OKPS + (size_t)sb * 42;           // (21,2) per sb
  const float* w = ws + (size_t)sb * NK * 4;
  float k0x = 0.f, k0y = 0.f;
  #pragma unroll
  for (int g = 0; g < 5; ++g) {                       // k = 0,4,8,12,16
    k0x += w[(4 * g) * 4 + 2];
    k0y += w[(4 * g) * 4 + 3];
  }
  kp[0] = k0x * (4.0f / 5.0f);
  kp[1] = k0y * (4.0f / 5.0f);
  #pragma unroll
  for (int k = 0; k < NK; ++k) {
    const int idx = 4 * (k / 4) + 4 - (k % 4);        // kp_idx permutation
    kp[idx * 2 + 0] = w[k * 4 + 0] * 4.0f;
    kp[idx * 2 + 1] = w[k * 4 + 1] * 4.0f;
  }
}

extern "C" void kernel_launch(void* const* d_in, const int* in_sizes, int n_in,
                              void* d_out, int out_size, void* d_ws, size_t ws_size,
                              hipStream_t stream) {
  const float* x = (const float*)d_in[0];
  float* out = (float*)d_out;
  float* ws  = (float*)d_ws;                          // needs SB*NK*4 floats = 40 KB
  const int grid = SB * NK + SB * 15;                 // 2560 fused + 1920 copy blocks
  skel_fused<<<grid, 256, 0, stream>>>(x, out, ws);
  skel_finalize<<<1, SB, 0, stream>>>(ws, out);
}